// MultiHeadAttention_49813030699558
// MI455X (gfx1250) — compile-verified
//
#include <hip/hip_runtime.h>
#include <hip/hip_bf16.h>
#include <math.h>

// ---------------------------------------------------------------------------
// Problem constants (from reference): B=2, L=2048, C=1024, H=16, D=64
// ---------------------------------------------------------------------------
#define BB     2
#define LL     2048
#define CC     1024
#define HH     16
#define DD     64
#define M_ROWS (BB * LL)      // 4096
#define QKV_N  (3 * CC)       // 3072

typedef __attribute__((ext_vector_type(2))) float v2f;
typedef __attribute__((ext_vector_type(8))) float v8f;

// ---------------------------------------------------------------------------
// CDNA5 async global->LDS copy (ASYNCcnt path, no VGPR round-trip).
// LDS destination = low 32 bits of the flat pointer (LDS aperture keeps the
// byte offset in addr[31:0]). Per-lane, 16B per lane per issue.
// ---------------------------------------------------------------------------
__device__ __forceinline__ void async_load_b128(void* lds_dst, const void* gsrc) {
  unsigned lds_off = (unsigned)(uintptr_t)lds_dst;
  asm volatile("global_load_async_to_lds_b128 %0, %1, off"
               :: "v"(lds_off), "v"(gsrc)
               : "memory");
}
__device__ __forceinline__ void wait_async_all() {
  asm volatile("s_wait_asynccnt 0x0" ::: "memory");
}

// ===========================================================================
// Kernel 1/4: C[M,N] = A[M,K] @ W[N,K]^T + bias[N]   (fp32, WMMA 16x16x4)
// 256 threads = 8 waves; tile 256x64; each wave owns 32 rows (2 A-frags)
// sharing 4 B-frags. Tiles staged via GLOBAL_LOAD_ASYNC_TO_LDS_B128.
// ===========================================================================
__global__ __launch_bounds__(256)
void gemm_bias_wmma(const float* __restrict__ A, const float* __restrict__ W,
                    const float* __restrict__ bias, float* __restrict__ Cout,
                    int M, int N, int K) {
  constexpr int TM = 256, TN = 64, TK = 32, LDT = TK + 4;  // +4: 16B-aligned, bank-safe
  __shared__ float As[TM * LDT];   // 36,864 B
  __shared__ float Ws[TN * LDT];   //  9,216 B

  const int tid  = threadIdx.x;
  const int wave = tid >> 5;
  const int lane = tid & 31;
  const int hh   = lane >> 4;   // half-wave: selects K pair {0,1} vs {2,3}
  const int l16  = lane & 15;
  const int m0 = blockIdx.y * TM;
  const int n0 = blockIdx.x * TN;

  v8f acc[2][4] = {};

  for (int k0 = 0; k0 < K; k0 += TK) {
    // A tile: 256x32 floats = 2048 float4 slots, 8 async b128 per thread
#pragma unroll
    for (int i = 0; i < 8; ++i) {
      int slot = tid + i * 256;
      int r = slot >> 3, c4 = slot & 7;
      async_load_b128(&As[r * LDT + c4 * 4],
                      A + (size_t)(m0 + r) * K + k0 + c4 * 4);
    }
    // W tile: 64x32 floats = 512 float4 slots, 2 async b128 per thread
#pragma unroll
    for (int i = 0; i < 2; ++i) {
      int slot = tid + i * 256;
      int r = slot >> 3, c4 = slot & 7;
      async_load_b128(&Ws[r * LDT + c4 * 4],
                      W + (size_t)(n0 + r) * K + k0 + c4 * 4);
    }
    wait_async_all();
    __syncthreads();

#pragma unroll
    for (int kk = 0; kk < TK; kk += 4) {
      // Group ALL fragment loads so ds_loads clause and one dscnt wait
      // covers 8 WMMAs.
      v2f a0, a1, bf[4];
      {
        const int r0 = wave * 32 + l16;
        a0.x = As[r0 * LDT + kk + hh * 2];
        a0.y = As[r0 * LDT + kk + hh * 2 + 1];
        const int r1 = r0 + 16;
        a1.x = As[r1 * LDT + kk + hh * 2];
        a1.y = As[r1 * LDT + kk + hh * 2 + 1];
      }
#pragma unroll
      for (int nt = 0; nt < 4; ++nt) {
        const int bcol = nt * 16 + l16;
        bf[nt].x = Ws[bcol * LDT + kk + hh * 2];
        bf[nt].y = Ws[bcol * LDT + kk + hh * 2 + 1];
      }
#pragma unroll
      for (int nt = 0; nt < 4; ++nt) {
        acc[0][nt] = __builtin_amdgcn_wmma_f32_16x16x4_f32(
            false, a0, false, bf[nt], (short)0, acc[0][nt], false, false);
        acc[1][nt] = __builtin_amdgcn_wmma_f32_16x16x4_f32(
            false, a1, false, bf[nt], (short)0, acc[1][nt], false, false);
      }
    }
    __syncthreads();
  }

  // Epilogue: C/D layout -> VGPR r holds row r (lanes 0-15) / r+8 (lanes 16-31)
#pragma unroll
  for (int rf = 0; rf < 2; ++rf) {
#pragma unroll
    for (int nt = 0; nt < 4; ++nt) {
      const int gn = n0 + nt * 16 + l16;
      const float bv = bias[gn];
#pragma unroll
      for (int r = 0; r < 8; ++r) {
        const int gm = m0 + wave * 32 + rf * 16 + hh * 8 + r;
        Cout[(size_t)gm * N + gn] = acc[rf][nt][r] + bv;
      }
    }
  }
}

// ===========================================================================
// Kernel 2: RoPE + QK-RMSNorm, in place on q/k slices of qkv [4096, 3072].
// One wave per (b,l,h); lane owns rotation pair p=lane (D/2 = 32 pairs).
// ===========================================================================
__global__ __launch_bounds__(256)
void rope_rmsnorm(float* __restrict__ qkv, const float* __restrict__ qg,
                  const float* __restrict__ kg) {
  const int wave = threadIdx.x >> 5;
  const int lane = threadIdx.x & 31;
  const int task = blockIdx.x * 8 + wave;        // over B*L*H
  const int h  = task % HH;
  const int bl = task / HH;                      // b*L + l
  const int l  = bl % LL;

  float* qp = qkv + (size_t)bl * QKV_N + h * DD;
  float* kp = qp + CC;

  const float inv_freq = powf(10000.0f, -(float)lane / 32.0f);
  const float ang = (float)l * inv_freq;
  const float c = cosf(ang), s = sinf(ang);
  const float g1q = qg[h * DD + 2 * lane], g2q = qg[h * DD + 2 * lane + 1];
  const float g1k = kg[h * DD + 2 * lane], g2k = kg[h * DD + 2 * lane + 1];

  // --- q ---
  {
    float t1 = qp[2 * lane], t2 = qp[2 * lane + 1];
    float r1 = t1 * c - t2 * s;
    float r2 = t1 * s + t2 * c;
    float ss = r1 * r1 + r2 * r2;
#pragma unroll
    for (int off = 16; off >= 1; off >>= 1) ss += __shfl_xor(ss, off, 32);
    float rinv = rsqrtf(ss * (1.0f / DD) + 1e-6f);
    qp[2 * lane]     = r1 * rinv * g1q;
    qp[2 * lane + 1] = r2 * rinv * g2q;
  }
  // --- k ---
  {
    float t1 = kp[2 * lane], t2 = kp[2 * lane + 1];
    float r1 = t1 * c - t2 * s;
    float r2 = t1 * s + t2 * c;
    float ss = r1 * r1 + r2 * r2;
#pragma unroll
    for (int off = 16; off >= 1; off >>= 1) ss += __shfl_xor(ss, off, 32);
    float rinv = rsqrtf(ss * (1.0f / DD) + 1e-6f);
    kp[2 * lane]     = r1 * rinv * g1k;
    kp[2 * lane + 1] = r2 * rinv * g2k;
  }
}

// ===========================================================================
// Kernel 3: flash attention per (b, h, 128-row Q block), fp32 WMMA.
// 256 threads = 8 waves; each wave owns 16 query rows. Key blocks of 64.
// K/V tiles staged with GLOBAL_LOAD_ASYNC_TO_LDS_B128 (ASYNCcnt).
// P (softmaxed scores) overlays the K LDS region (K is dead after S-compute).
// ===========================================================================
__global__ __launch_bounds__(256)
void flash_attn(const float* __restrict__ qkv, float* __restrict__ obuf) {
  constexpr int LDK = DD + 4;   // 68: 16B-aligned rows, bank-conflict free
  constexpr int LDP = DD + 2;   // 66: scalar access only, conflict free
  __shared__ float smemKP[8 * 16 * LDP];   // 8448 floats: Ks (4352) then Ps
  __shared__ float smemV[64 * LDK];        // 4352 floats

  const int tid  = threadIdx.x;
  const int wave = tid >> 5;
  const int lane = tid & 31;
  const int hh   = lane >> 4;
  const int l16  = lane & 15;

  const int nmb = LL / 128;                // 16 Q blocks per (b,h)
  const int mb  = blockIdx.x % nmb;
  const int bh  = blockIdx.x / nmb;
  const int h   = bh % HH;
  const int b   = bh / HH;

  const float* qbase = qkv + (size_t)b * LL * QKV_N + h * DD;          // q slice
  const float* kbase = qbase + CC;
  const float* vbase = qbase + 2 * CC;
  const int m0 = mb * 128;

  // Preload this wave's Q A-fragments (row = wave*16 + l16; half picks K pair)
  float qreg[32];
  {
    const float* qrow = qbase + (size_t)(m0 + wave * 16 + l16) * QKV_N;
#pragma unroll
    for (int s = 0; s < 16; ++s) {
      qreg[2 * s]     = qrow[4 * s + hh * 2];
      qreg[2 * s + 1] = qrow[4 * s + hh * 2 + 1];
    }
  }

  v8f oacc[4] = {};
  float mrow[8], lrow[8];
#pragma unroll
  for (int r = 0; r < 8; ++r) { mrow[r] = -1e30f; lrow[r] = 0.0f; }
  const float scale = 0.125f;   // 1/sqrt(64)
  float* Ks = smemKP;                       // stride LDK, 64 rows
  float* Pw = smemKP + wave * 16 * LDP;     // per-wave 16x64, stride LDP

  for (int j0 = 0; j0 < LL; j0 += 64) {
    // Async-stage K, V tiles: 64 rows x 64 floats each = 1024 b128 slots,
    // 4 per thread per tile, straight into LDS.
#pragma unroll
    for (int i = 0; i < 4; ++i) {
      int slot = tid + i * 256;
      int r = slot >> 4, c4 = slot & 15;
      async_load_b128(&Ks[r * LDK + c4 * 4],
                      kbase + (size_t)(j0 + r) * QKV_N + c4 * 4);
      async_load_b128(&smemV[r * LDK + c4 * 4],
                      vbase + (size_t)(j0 + r) * QKV_N + c4 * 4);
    }
    wait_async_all();
    __syncthreads();

    // S = Q @ K^T  (16x64 per wave); group B-frag loads before the WMMAs
    v8f sacc[4] = {};
#pragma unroll
    for (int s = 0; s < 16; ++s) {
      v2f a; a.x = qreg[2 * s]; a.y = qreg[2 * s + 1];
      const int kk = 4 * s;
      v2f bf[4];
#pragma unroll
      for (int nt = 0; nt < 4; ++nt) {
        bf[nt].x = Ks[(nt * 16 + l16) * LDK + kk + hh * 2];
        bf[nt].y = Ks[(nt * 16 + l16) * LDK + kk + hh * 2 + 1];
      }
#pragma unroll
      for (int nt = 0; nt < 4; ++nt)
        sacc[nt] = __builtin_amdgcn_wmma_f32_16x16x4_f32(
            false, a, false, bf[nt], (short)0, sacc[nt], false, false);
    }
    __syncthreads();   // all waves done reading Ks; safe to overlay with P

    // Online softmax (row reductions within 16-lane groups match C layout)
#pragma unroll
    for (int r = 0; r < 8; ++r) {
      float v = -1e30f;
#pragma unroll
      for (int nt = 0; nt < 4; ++nt) v = fmaxf(v, sacc[nt][r] * scale);
#pragma unroll
      for (int off = 8; off >= 1; off >>= 1) v = fmaxf(v, __shfl_xor(v, off, 32));
      const float mnew = fmaxf(mrow[r], v);
      const float alpha = __expf(mrow[r] - mnew);
      mrow[r] = mnew;
      float rs = 0.0f;
#pragma unroll
      for (int nt = 0; nt < 4; ++nt) {
        float p = __expf(sacc[nt][r] * scale - mnew);
        sacc[nt][r] = p;
        rs += p;
      }
#pragma unroll
      for (int off = 8; off >= 1; off >>= 1) rs += __shfl_xor(rs, off, 32);
      lrow[r] = lrow[r] * alpha + rs;
#pragma unroll
      for (int nt = 0; nt < 4; ++nt) oacc[nt][r] *= alpha;
    }

    // P: C layout -> LDS (per-wave buffer, overlaying Ks region)
#pragma unroll
    for (int nt = 0; nt < 4; ++nt)
#pragma unroll
      for (int r = 0; r < 8; ++r)
        Pw[(hh * 8 + r) * LDP + nt * 16 + l16] = sacc[nt][r];
    __syncthreads();

    // O += P @ V  (K dim = 64 keys); grouped fragment loads
#pragma unroll
    for (int s = 0; s < 16; ++s) {
      const int kk = 4 * s;
      v2f a, bf[4];
      a.x = Pw[l16 * LDP + kk + hh * 2];
      a.y = Pw[l16 * LDP + kk + hh * 2 + 1];
#pragma unroll
      for (int nt = 0; nt < 4; ++nt) {
        bf[nt].x = smemV[(kk + hh * 2) * LDK + nt * 16 + l16];
        bf[nt].y = smemV[(kk + hh * 2 + 1) * LDK + nt * 16 + l16];
      }
#pragma unroll
      for (int nt = 0; nt < 4; ++nt)
        oacc[nt] = __builtin_amdgcn_wmma_f32_16x16x4_f32(
            false, a, false, bf[nt], (short)0, oacc[nt], false, false);
    }
    __syncthreads();   // before next K/V tile load (and P overlay safety)
  }

  // Epilogue: O / l -> obuf[(b*L + m), h*64 + n]  (blhd layout, [4096,1024])
#pragma unroll
  for (int r = 0; r < 8; ++r) {
    const float inv = 1.0f / lrow[r];
    const int gm = b * LL + m0 + wave * 16 + hh * 8 + r;
#pragma unroll
    for (int nt = 0; nt < 4; ++nt)
      obuf[(size_t)gm * CC + h * DD + nt * 16 + l16] = oacc[nt][r] * inv;
  }
}

// ===========================================================================
// Launch: qkv GEMM -> RoPE/RMSNorm -> flash attention -> output GEMM
// Workspace: qkv (4096x3072 f32 = 48MB) + obuf (4096x1024 f32 = 16MB)
// ===========================================================================
extern "C" void kernel_launch(void* const* d_in, const int* in_sizes, int n_in,
                              void* d_out, int out_size, void* d_ws, size_t ws_size,
                              hipStream_t stream) {
  (void)in_sizes; (void)n_in; (void)out_size; (void)ws_size;
  const float* x       = (const float*)d_in[0];
  const float* w_qkv   = (const float*)d_in[1];
  const float* b_qkv   = (const float*)d_in[2];
  const float* w_out   = (const float*)d_in[3];
  const float* b_out   = (const float*)d_in[4];
  const float* q_gamma = (const float*)d_in[5];
  const float* k_gamma = (const float*)d_in[6];

  float* qkv  = (float*)d_ws;
  float* obuf = qkv + (size_t)M_ROWS * QKV_N;
  float* out  = (float*)d_out;

  dim3 g1(QKV_N / 64, M_ROWS / 256);   // 48 x 16
  gemm_bias_wmma<<<g1, 256, 0, stream>>>(x, w_qkv, b_qkv, qkv,
                                         M_ROWS, QKV_N, CC);

  rope_rmsnorm<<<(BB * LL * HH) / 8, 256, 0, stream>>>(qkv, q_gamma, k_gamma);

  flash_attn<<<BB * HH * (LL / 128), 256, 0, stream>>>(qkv, obuf);

  dim3 g4(CC / 64, M_ROWS / 256);      // 16 x 16
  gemm_bias_wmma<<<g4, 256, 0, stream>>>(obuf, w_out, b_out, out,
                                         M_ROWS, CC, CC);
}